// LabelSmoothingCTCLoss_86801289052897
// MI455X (gfx1250) — compile-verified
//
#include <hip/hip_runtime.h>
#include <stdint.h>

// Problem constants (from reference)
#define T_DIM   2048
#define B_DIM   96
#define C_DIM   128
#define L_DIM   128
#define S_DIM   (2 * L_DIM + 1)      // 257 extended states
#define CHUNK   8                    // time rows per TDM tile
#define NCHUNK  (T_DIM / CHUNK)      // 256 chunks
#define NBUF    3                    // triple-buffered tiles, prefetch depth 2
#define NEGF    (-1e30f)

typedef __attribute__((ext_vector_type(4))) unsigned int v4u;
typedef __attribute__((ext_vector_type(8))) int          v8i;
typedef __attribute__((ext_vector_type(4))) int          v4i;

// Issue a TDM load of CHUNK rows x C_DIM f32 (row stride B*C elements) from
// gsrc into LDS at byte offset lds_off. D# layout per CDNA5 ISA ch. 8.
// This toolchain (clang-23 / therock-10.0 HIP headers) exposes the 6-arg form:
//   (uint32x4 g0, int32x8 g1, int32x4 g2, int32x4 g3, int32x8 g4, i32 cpol)
__device__ __forceinline__ void tdm_load_rows(const float* gsrc, unsigned lds_off) {
    unsigned long long ga = (unsigned long long)(uintptr_t)gsrc;
    v4u g0;
    g0.x = 1u;                                              // count=1 (valid), user mode
    g0.y = lds_off;                                         // lds_addr (bytes)
    g0.z = (unsigned)(ga & 0xFFFFFFFFull);                  // global_addr[31:0]
    g0.w = (unsigned)((ga >> 32) & 0x01FFFFFFull)           // global_addr[56:32]
         | (2u << 30);                                      // type = 2 ("image")
    v8i g1;
    g1[0] = (int)(2u << 16);                                // data_size = 2 -> 4 bytes
    g1[1] = (int)((unsigned)(C_DIM & 0xFFFF) << 16);        // tensor_dim0 lo16 (=128)
    g1[2] = (int)(((unsigned)T_DIM & 0xFFFFu) << 16);       // dim0 hi16=0 | tensor_dim1 lo16 (=2048)
    g1[3] = (int)((unsigned)C_DIM << 16);                   // dim1 hi16=0 | tile_dim0 = 128
    g1[4] = CHUNK;                                          // tile_dim1 = 8 | tile_dim2 = 0
    g1[5] = B_DIM * C_DIM;                                  // tensor_dim0_stride lo32 (12288 elems)
    g1[6] = 0;                                              // stride hi | dim1_stride lo
    g1[7] = 0;
    v4i g2 = {0, 0, 0, 0};                                  // unused (2-D tensor)
    v4i g3 = {0, 0, 0, 0};                                  // unused (2-D tensor)
    v8i g4 = {0, 0, 0, 0, 0, 0, 0, 0};                      // unused trailing group
    __builtin_amdgcn_tensor_load_to_lds(g0, g1, g2, g3, g4, 0);
}

__global__ void __launch_bounds__(256)
ctc_forward_kernel(const float* __restrict__ log_probs,
                   const int*   __restrict__ targets,
                   const int*   __restrict__ input_lengths,
                   const int*   __restrict__ target_lengths,
                   float*       __restrict__ per_b_loss)
{
    __shared__ float         lp_tile[NBUF][CHUNK * C_DIM]; // triple-buffered TDM tiles (12 KB)
    __shared__ float         alphaA[S_DIM + 3];
    __shared__ float         alphaB[S_DIM + 3];
    __shared__ int           ext_lab[S_DIM];
    __shared__ unsigned char cskip[S_DIM];
    __shared__ float         red[256];

    const int tid  = threadIdx.x;
    const int b    = blockIdx.x;
    const int ilen = input_lengths[b];
    const int tlen = target_lengths[b];

    // Extended labels: blank, y1, blank, y2, ..., yL, blank; skip allowed for
    // odd s>=3 when the label differs from the label two states back.
    for (int s = tid; s < S_DIM; s += 256) {
        int e = (s & 1) ? targets[b * L_DIM + ((s - 1) >> 1)] : 0;
        ext_lab[s] = e;
        unsigned char cs = 0;
        if ((s & 1) && s >= 3)
            cs = (unsigned char)(targets[b * L_DIM + ((s - 1) >> 1)] !=
                                 targets[b * L_DIM + ((s - 3) >> 1)]);
        cskip[s] = cs;
    }

    const float* gb = log_probs + (size_t)b * C_DIM;  // start of [T, C] slice for batch b

    // Wave 0 prefetches chunks 0 and 1 while everyone builds ext/cskip.
    if (tid < 32) {
        tdm_load_rows(gb, (unsigned)(uintptr_t)&lp_tile[0][0]);
        tdm_load_rows(gb + (size_t)CHUNK * (B_DIM * C_DIM),
                      (unsigned)(uintptr_t)&lp_tile[1][0]);
    }

    float  kacc = 0.0f;       // sum of log-probs over (t<ilen, c!=blank) for KL term
    float* cur  = alphaA;
    float* nxt  = alphaB;

    int buf = 0;              // k % NBUF, tracked incrementally
    for (int k = 0; k < NCHUNK; ++k) {
        if (tid < 32) {
            if (k + 2 < NCHUNK) {
                // Prefetch chunk k+2 (depth-2 pipeline), then wait until chunk k
                // has landed. In-order per-wave TDM completion: after issuing
                // k+2 there are <=3 outstanding (k, k+1, k+2); cnt<=2 => k done.
                int nb = buf + 2; if (nb >= NBUF) nb -= NBUF;
                tdm_load_rows(gb + (size_t)(k + 2) * CHUNK * (B_DIM * C_DIM),
                              (unsigned)(uintptr_t)&lp_tile[nb][0]);
                __builtin_amdgcn_s_wait_tensorcnt(2);
            } else if (k + 1 < NCHUNK) {
                __builtin_amdgcn_s_wait_tensorcnt(1);   // only k, k+1 outstanding
            } else {
                __builtin_amdgcn_s_wait_tensorcnt(0);   // last chunk
            }
        }
        __syncthreads();  // chunk k visible to all waves (and ext/cskip on k==0)

        const int t0 = k * CHUNK;

        // Fused label-smoothing KL accumulation over this tile.
        for (int i = tid; i < CHUNK * C_DIM; i += 256) {
            int c = i & (C_DIM - 1);
            int t = t0 + (i >> 7);
            if (c != 0 && t < ilen) kacc += lp_tile[buf][i];
        }

        // Sequential alpha recursion over the CHUNK time rows.
        for (int r = 0; r < CHUNK; ++r) {
            const int t = t0 + r;
            const float* lp_row = &lp_tile[buf][r * C_DIM];
            if (t == 0) {
                for (int s = tid; s < S_DIM; s += 256)
                    cur[s] = (s <= 1) ? lp_row[ext_lab[s]] : NEGF;
            } else {
                for (int s = tid; s < S_DIM; s += 256) {
                    float a0 = cur[s];
                    float a1 = (s >= 1) ? cur[s - 1] : NEGF;
                    float a2 = (s >= 2 && cskip[s]) ? cur[s - 2] : NEGF;
                    float m  = fmaxf(a0, fmaxf(a1, a2));
                    float lse = m + __logf(__expf(a0 - m) + __expf(a1 - m) + __expf(a2 - m));
                    float nv  = lse + lp_row[ext_lab[s]];
                    nxt[s] = (t < ilen) ? nv : a0;   // freeze past each sample's length
                }
            }
            __syncthreads();
            if (t != 0) { float* tmp = cur; cur = nxt; nxt = tmp; }
        }
        // The barrier after row r=CHUNK-1 ends all reads of lp_tile[buf]; the
        // next write into this buffer (chunk k+3) is only issued after it.

        if (++buf >= NBUF) buf = 0;
    }

    // Block-wide reduction of the KL accumulator.
    red[tid] = kacc;
    __syncthreads();
    for (int off = 128; off > 0; off >>= 1) {
        if (tid < off) red[tid] += red[tid + off];
        __syncthreads();
    }

    if (tid == 0) {
        float a1 = cur[2 * tlen];
        float a2 = cur[2 * tlen - 1];
        float m  = fmaxf(a1, a2);
        float ll = m + __logf(__expf(a1 - m) + __expf(a2 - m));
        float ctc = (ll < -5e29f) ? 0.0f : -ll;   // zero_infinity
        const float u = 1.0f / (float)(C_DIM - 1);
        float klm = __logf(u) - u * red[0] / (float)ilen;
        per_b_loss[b] = 0.9f * ctc + 0.1f * klm;  // (1-SMOOTHING)*ctc + SMOOTHING*kl
    }
}

__global__ void __launch_bounds__(128)
finalize_kernel(const float* __restrict__ per_b, float* __restrict__ out)
{
    __shared__ float red[128];
    const int tid = threadIdx.x;
    red[tid] = (tid < B_DIM) ? per_b[tid] : 0.0f;
    __syncthreads();
    for (int off = 64; off > 0; off >>= 1) {
        if (tid < off) red[tid] += red[tid + off];
        __syncthreads();
    }
    if (tid == 0) out[0] = red[0] / (float)B_DIM;
}

extern "C" void kernel_launch(void* const* d_in, const int* in_sizes, int n_in,
                              void* d_out, int out_size, void* d_ws, size_t ws_size,
                              hipStream_t stream) {
    (void)in_sizes; (void)n_in; (void)out_size; (void)ws_size;
    const float* log_probs      = (const float*)d_in[0];
    const int*   targets        = (const int*)d_in[1];
    const int*   input_lengths  = (const int*)d_in[2];
    const int*   target_lengths = (const int*)d_in[3];
    float*       per_b          = (float*)d_ws;

    ctc_forward_kernel<<<B_DIM, 256, 0, stream>>>(log_probs, targets, input_lengths,
                                                  target_lengths, per_b);
    finalize_kernel<<<1, 128, 0, stream>>>(per_b, (float*)d_out);
}